// MoEBlock_74105365725734
// MI455X (gfx1250) — compile-verified
//
#include <hip/hip_runtime.h>
#include <math.h>

// ---------------- problem constants ----------------
#define B_  16
#define N_  1024
#define C_  768
#define E_  8
#define H_  192
#define D_  4
#define K_  2
#define E2_ 16            // 2*E

typedef float v2f __attribute__((ext_vector_type(2)));
typedef float v8f __attribute__((ext_vector_type(8)));
typedef float v4f __attribute__((ext_vector_type(4)));
typedef int   v4i __attribute__((ext_vector_type(4)));

#define AS1 __attribute__((address_space(1)))
#define AS3 __attribute__((address_space(3)))

#if __has_builtin(__builtin_amdgcn_global_load_async_to_lds_b32) && \
    __has_builtin(__builtin_amdgcn_global_load_async_to_lds_b128) && \
    __has_builtin(__builtin_amdgcn_s_wait_asynccnt)
#define HAVE_ASYNC_LDS 1
#else
#define HAVE_ASYNC_LDS 0
#endif

// ---------------- LDS layout for expert kernel ----------------
// H_s  : [192][68]  transposed activations (pitch 68 -> rows r and r+8 hit
//                   different bank groups for the half-wave store pattern)
// arena: union of { Xt[32][65] + W1s[192][36] }  (stage 1)
//                 { W2s[128][36] }               (stage 2)
// pitch 36 (mult. of 4) keeps b128 LDS destinations 16B-aligned and
// 36*h mod 64 is injective over h=0..15 -> conflict-free A-fragment reads.
#define HS_PITCH   68
#define HS_FLOATS  (H_ * HS_PITCH)            // 13056
#define XT_PITCH   65
#define XT_FLOATS  (32 * XT_PITCH)            // 2080
#define W1_PITCH   36
#define W1_FLOATS  (H_ * W1_PITCH)            // 6912
#define W2_PITCH   36
#define ARENA_FLOATS (XT_FLOATS + W1_FLOATS)  // 8992
#define SMEM_FLOATS  (HS_FLOATS + ARENA_FLOATS)
#define SMEM_BYTES   (SMEM_FLOATS * 4)        // 88192

__device__ __forceinline__ v8f wmma_f32_k4(v2f a, v2f b, v8f c) {
    return __builtin_amdgcn_wmma_f32_16x16x4_f32(
        /*neg_a=*/false, a, /*neg_b=*/false, b,
        /*c_mod=*/(short)0, c, /*reuse_a=*/false, /*reuse_b=*/false);
}

#if HAVE_ASYNC_LDS
__device__ __forceinline__ void async_b32(const float* g, float* l) {
    __builtin_amdgcn_global_load_async_to_lds_b32(
        (AS1 int*)g, (AS3 int*)l, 0, 0);
}
__device__ __forceinline__ void async_b128(const float* g, float* l) {
    __builtin_amdgcn_global_load_async_to_lds_b128(
        (AS1 v4i*)g, (AS3 v4i*)l, 0, 0);
}
#endif

// ============ kernel 0: out = x (float4), zero gate accumulators ============
__global__ void moe_init_kernel(const v4f* __restrict__ x4,
                                v4f* __restrict__ out4,
                                float* __restrict__ exp_sum) {
    int gid = blockIdx.x * 256 + threadIdx.x;
    out4[gid] = x4[gid];                        // grid sized exactly B*N*C/4
    if (gid < B_ * E_) exp_sum[gid] = 0.0f;
}

// ============ kernel 1: noisy gate logits, sum over N ============
// grid (64 ntiles, 16 b), 256 threads: 16 rows x 16 gate outputs
__global__ void moe_gate_kernel(const float* __restrict__ x,
                                const int*   __restrict__ task_ids,
                                const float* __restrict__ eps,
                                const float* __restrict__ gate_w,
                                float* __restrict__ exp_sum) {
    __shared__ float s_acc[16][E2_ + 1];
    __shared__ float s_red[E_][17];

    const int bb = blockIdx.y;
    const int n0 = blockIdx.x * 16;
    const int t  = threadIdx.x;
    const int nl = t >> 4;           // 0..15
    const int e2 = t & 15;           // 0..15

    const int task = task_ids[bb];
    const float* xr = x + ((size_t)(bb * N_ + n0 + nl)) * C_;
    const float* gw = gate_w + (size_t)task * C_ * E2_ + e2;

    float acc = 0.0f;
#pragma unroll 4
    for (int c = 0; c < C_; ++c)
        acc = fmaf(xr[c], gw[(size_t)c * E2_], acc);
    s_acc[nl][e2] = acc;
    __syncthreads();

    if (t < 128) {
        const int n = t >> 3, e = t & 7;
        float clean = s_acc[n][e];
        float raw   = s_acc[n][e + 8];
        float sp    = (raw > 20.0f) ? raw : log1pf(expf(raw));  // softplus
        float sd    = sp + 0.01f;
        float lg    = clean + eps[((size_t)(bb * N_ + n0 + n)) * E_ + e] * sd;
        s_red[e][n] = lg;
    }
    __syncthreads();
    if (t < E_) {
        float s = 0.0f;
#pragma unroll
        for (int n = 0; n < 16; ++n) s += s_red[t][n];
        atomicAdd(exp_sum + bb * E_ + t, s);
    }
}

// ============ kernel 2: top-2 + rescale + softmax ============
__global__ void moe_topk_kernel(const float* __restrict__ exp_sum,
                                int* __restrict__ sel_idx,
                                float* __restrict__ sel_gate) {
    int b = threadIdx.x;
    if (b >= B_) return;
    float v[E_];
#pragma unroll
    for (int e = 0; e < E_; ++e) v[e] = exp_sum[b * E_ + e];

    int i0 = 0;
#pragma unroll
    for (int e = 1; e < E_; ++e) if (v[e] > v[i0]) i0 = e;
    int i1 = (i0 == 0) ? 1 : 0;
#pragma unroll
    for (int e = 0; e < E_; ++e) if (e != i0 && v[e] > v[i1]) i1 = e;

    // top2 desc: (v[i0], v[i1]); scaled = (vi - m1)/(m2 - m1 + 1e-6)
    float d  = v[i0] - v[i1];
    float s0 = d / (d + 1e-6f);                 // scaled1 == 0
    float z  = expf(-s0);
    float g0 = 1.0f / (1.0f + z);
    float g1 = z / (1.0f + z);

    sel_idx[b * K_ + 0] = i0;  sel_gate[b * K_ + 0] = g0;
    sel_idx[b * K_ + 1] = i1;  sel_gate[b * K_ + 1] = g1;
}

// ============ kernel 3: sparse expert FFN via f32 WMMA ============
// grid (16 row-tiles of 64, 16 b), 256 threads = 8 waves
__global__ void moe_expert_kernel(const float* __restrict__ x,
                                  const float* __restrict__ fc1_w,
                                  const float* __restrict__ fc1_b,
                                  const float* __restrict__ fc2_w,
                                  const float* __restrict__ fc2_b,
                                  const int*   __restrict__ sel_idx,
                                  const float* __restrict__ sel_gate,
                                  float* __restrict__ out) {
    extern __shared__ float smem[];
    float* H_s   = smem;                 // [192][68]
    float* arena = smem + HS_FLOATS;
    float* Xt    = arena;                // [32][65]  (stage 1)
    float* W1s   = arena + XT_FLOATS;    // [192][36] (stage 1)
    float* W2s   = arena;                // [128][36] (stage 2)

    const int m0   = blockIdx.x * 64;    // row-tile base (n)
    const int bb   = blockIdx.y;
    const int tid  = threadIdx.x;
    const int wave = tid >> 5;
    const int lane = tid & 31;
    const int ln   = lane & 15;          // col within tile
    const int kh   = (lane >> 4) << 1;   // K offset 0 / 2 per half-wave
    const int mhi  = (lane >> 4) << 3;   // row  +0 / +8 per half-wave

    const float* xblk = x + ((size_t)(bb * N_ + m0)) * C_;

    for (int k = 0; k < K_; ++k) {
        const int   e    = sel_idx[bb * K_ + k];
        const float gate = sel_gate[bb * K_ + k];
        const float* w1e = fc1_w + (size_t)e * H_ * C_;
        const float* b1e = fc1_b + (size_t)e * H_;
        const float* w2e = fc2_w + (size_t)e * C_ * H_;
        const float* b2e = fc2_b + (size_t)e * C_;

        // ---------- stage 1: D[h,n] = W1 * X^T over C, chunks of 32 ----------
        v8f acc1[6];
#pragma unroll
        for (int tl = 0; tl < 6; ++tl)
#pragma unroll
            for (int v = 0; v < 8; ++v) acc1[tl][v] = 0.0f;

        for (int cc = 0; cc < C_; cc += 32) {
            __syncthreads();
#if HAVE_ASYNC_LDS
            // transpose x chunk: per-lane LDS dest, b32 async (8/thread)
#pragma unroll
            for (int i = 0; i < 8; ++i) {
                int idx = tid + i * 256;
                int n = idx >> 5, c = idx & 31;
                async_b32(xblk + (size_t)n * C_ + cc + c,
                          Xt + c * XT_PITCH + n);
            }
            // W1 chunk rows are contiguous: b128 async (6/thread)
#pragma unroll
            for (int i = 0; i < 6; ++i) {
                int idx = tid + i * 256;
                int h = idx >> 3, c4 = (idx & 7) << 2;
                async_b128(w1e + (size_t)h * C_ + cc + c4,
                           W1s + h * W1_PITCH + c4);
            }
            __builtin_amdgcn_s_wait_asynccnt(0);
#else
#pragma unroll
            for (int i = 0; i < 8; ++i) {
                int idx = tid + i * 256;
                int n = idx >> 5, c = idx & 31;
                Xt[c * XT_PITCH + n] = xblk[(size_t)n * C_ + cc + c];
            }
#pragma unroll
            for (int i = 0; i < 24; ++i) {
                int idx = tid + i * 256;
                int h = idx >> 5, c = idx & 31;
                W1s[h * W1_PITCH + c] = w1e[(size_t)h * C_ + cc + c];
            }
#endif
            __syncthreads();

#pragma unroll
            for (int tl = 0; tl < 6; ++tl) {
                const int t  = wave * 6 + tl;
                const int mh = t >> 2;        // h-tile 0..11
                const int nt = t & 3;         // n-tile 0..3
                const float* arow = W1s + (mh * 16 + ln) * W1_PITCH;
#pragma unroll
                for (int s = 0; s < 32; s += 4) {
                    v2f a, b;
                    a[0] = arow[s + kh];
                    a[1] = arow[s + kh + 1];
                    b[0] = Xt[(s + kh)     * XT_PITCH + nt * 16 + ln];
                    b[1] = Xt[(s + kh + 1) * XT_PITCH + nt * 16 + ln];
                    acc1[tl] = wmma_f32_k4(a, b, acc1[tl]);
                }
            }
        }
        __syncthreads();

        // bias + exact gelu, park H in LDS as [h][n]
#pragma unroll
        for (int tl = 0; tl < 6; ++tl) {
            const int t  = wave * 6 + tl;
            const int mh = t >> 2, nt = t & 3;
#pragma unroll
            for (int v = 0; v < 8; ++v) {
                int   hh  = mh * 16 + v + mhi;
                float val = acc1[tl][v] + b1e[hh];
                val = 0.5f * val * (1.0f + erff(val * 0.70710678118654752f));
                H_s[hh * HS_PITCH + nt * 16 + ln] = val;
            }
        }
        __syncthreads();

        // ---------- stage 2: D[c,n] = W2 * H^T over H, groups of 128 c ------
        for (int gq = 0; gq < 6; ++gq) {
            const int cbase = gq * 128;
            v8f acc2[4];
#pragma unroll
            for (int nt = 0; nt < 4; ++nt)
#pragma unroll
                for (int v = 0; v < 8; ++v) acc2[nt][v] = 0.0f;

            for (int hc = 0; hc < H_; hc += 32) {
#if HAVE_ASYNC_LDS
                // W2 chunk rows are contiguous: b128 async (4/thread)
#pragma unroll
                for (int i = 0; i < 4; ++i) {
                    int idx = tid + i * 256;
                    int r = idx >> 3, c4 = (idx & 7) << 2;
                    async_b128(w2e + (size_t)(cbase + r) * H_ + hc + c4,
                               W2s + r * W2_PITCH + c4);
                }
                __builtin_amdgcn_s_wait_asynccnt(0);
#else
#pragma unroll
                for (int i = 0; i < 16; ++i) {
                    int idx = tid + i * 256;
                    int r = idx >> 5, c = idx & 31;
                    W2s[r * W2_PITCH + c] = w2e[(size_t)(cbase + r) * H_ + hc + c];
                }
#endif
                __syncthreads();

                const float* arow = W2s + (wave * 16 + ln) * W2_PITCH;
#pragma unroll
                for (int s = 0; s < 32; s += 4) {
                    v2f a;
                    a[0] = arow[s + kh];
                    a[1] = arow[s + kh + 1];
#pragma unroll
                    for (int nt = 0; nt < 4; ++nt) {
                        v2f b;
                        b[0] = H_s[(hc + s + kh)     * HS_PITCH + nt * 16 + ln];
                        b[1] = H_s[(hc + s + kh + 1) * HS_PITCH + nt * 16 + ln];
                        acc2[nt] = wmma_f32_k4(a, b, acc2[nt]);
                    }
                }
                __syncthreads();
            }

            // gated accumulate into out (out was pre-initialized to x)
#pragma unroll
            for (int nt = 0; nt < 4; ++nt) {
                const int n = nt * 16 + ln;
                float* orow = out + ((size_t)(bb * N_ + m0 + n)) * C_;
#pragma unroll
                for (int v = 0; v < 8; ++v) {
                    int cm = cbase + wave * 16 + v + mhi;
                    atomicAdd(orow + cm, gate * (acc2[nt][v] + b2e[cm]));
                }
            }
        }
        __syncthreads();
    }
}

// ---------------- host-side launch ----------------
extern "C" void kernel_launch(void* const* d_in, const int* in_sizes, int n_in,
                              void* d_out, int out_size, void* d_ws, size_t ws_size,
                              hipStream_t stream) {
    const float* x       = (const float*)d_in[0];
    const int*   task_id = (const int*)  d_in[1];
    const float* eps     = (const float*)d_in[2];
    const float* gate_w  = (const float*)d_in[3];
    const float* fc1_w   = (const float*)d_in[4];
    const float* fc1_b   = (const float*)d_in[5];
    const float* fc2_w   = (const float*)d_in[6];
    const float* fc2_b   = (const float*)d_in[7];
    float* out = (float*)d_out;

    float* exp_sum  = (float*)d_ws;                 // B*E floats
    int*   sel_idx  = (int*)(exp_sum + B_ * E_);    // B*K ints
    float* sel_gate = (float*)(sel_idx + B_ * K_);  // B*K floats

    // 0) out = x (vectorized), zero accumulators
    moe_init_kernel<<<(B_ * N_ * C_ / 4) / 256, 256, 0, stream>>>(
        (const v4f*)x, (v4f*)out, exp_sum);

    // 1) noisy gate logits summed over N
    moe_gate_kernel<<<dim3(N_ / 16, B_), 256, 0, stream>>>(
        x, task_id, eps, gate_w, exp_sum);

    // 2) top-2 -> rescale -> softmax gates
    moe_topk_kernel<<<1, 32, 0, stream>>>(exp_sum, sel_idx, sel_gate);

    // 3) sparse 2-expert FFN with f32 WMMA, gated atomic accumulate
    moe_expert_kernel<<<dim3(N_ / 64, B_), 256, SMEM_BYTES, stream>>>(
        x, fc1_w, fc1_b, fc2_w, fc2_b, sel_idx, sel_gate, out);
}